// Conv2D_12094627905542
// MI455X (gfx1250) — compile-verified
//
#include <hip/hip_runtime.h>

typedef __attribute__((ext_vector_type(2))) float v2f;
typedef __attribute__((ext_vector_type(8))) float v8f;

#define IN_C   32
#define OUT_C  64
#define HH     224
#define WW     224
#define NB     16
#define CSTR   33                      // padded channel stride (conflict-free ds)
#define ICOLS  34                      // 32-wide segment + 2 halo columns
#define ROWS_PER_WG 8
#define IROWS  (ROWS_PER_WG + 2)       // 8 output rows + halo
#define WLDS_F (9 * OUT_C * CSTR)      // 19008 floats
#define ILDS_F (IROWS * ICOLS * CSTR)  // 11220 floats

#define WMMA_F32X4(A, B, C) \
    __builtin_amdgcn_wmma_f32_16x16x4_f32(false, (A), false, (B), (short)0, (C), false, false)

__global__ __launch_bounds__(256)
void conv2d_wmma_f32_kernel(const float* __restrict__ x,
                            const float* __restrict__ wgt,
                            const float* __restrict__ bias,
                            float* __restrict__ out)
{
    extern __shared__ float lds[];
    float* wlds = lds;                 // [t = kh*3+kw][oc][c]  stride CSTR
    float* ilds = lds + WLDS_F;        // [row][col][c]         stride CSTR

    const int tid  = threadIdx.x;
    const int lane = tid & 31;
    const int wave = tid >> 5;

    const int seg = blockIdx.x % 7;
    const int h0  = ((blockIdx.x / 7) % (HH / ROWS_PER_WG)) * ROWS_PER_WG;
    const int n   =  blockIdx.x / (7 * (HH / ROWS_PER_WG));
    const int w0  = seg * 32;

    // ---- stage weights OIHW -> LDS [t][oc][c]
    for (int i = tid; i < 9 * OUT_C * IN_C; i += 256) {
        int c  = i & 31;
        int oc = (i >> 5) & 63;
        int t  = i >> 11;                          // 0..8 = kh*3+kw
        wlds[(t * OUT_C + oc) * CSTR + c] = wgt[(oc * IN_C + c) * 9 + t];
    }

    // ---- stage input slab rows h0-1 .. h0+8, zero-padded halo: LDS [row][col][c]
    for (int i = tid; i < IROWS * ICOLS * IN_C; i += 256) {
        int pos = i % (IROWS * ICOLS);             // consecutive tids -> contiguous w
        int c   = i / (IROWS * ICOLS);
        int col = pos % ICOLS;
        int row = pos / ICOLS;
        int hin = h0 - 1 + row;
        int win = w0 - 1 + col;
        float v = 0.0f;
        if (hin >= 0 && hin < HH && win >= 0 && win < WW)
            v = x[((n * IN_C + c) * HH + hin) * WW + win];
        ilds[(row * ICOLS + col) * CSTR + c] = v;
    }

    __syncthreads();

    const int rp  = wave & 3;          // row pair: output rows 2*rp, 2*rp+1
    const int o   = wave >> 2;         // which 32-wide oc half
    const int g   = lane >> 4;         // half-wave group: selects K pair (ISA layout)
    const int l16 = lane & 15;

    // 2x4 register block: 8 independent accumulation chains
    // acc[a][s]: a = oc 16-block (0/1), s = {row0/row1} x {w lo/hi}
    v8f acc00, acc01, acc02, acc03;    // a=0: (r0,lo) (r0,hi) (r1,lo) (r1,hi)
    v8f acc10, acc11, acc12, acc13;    // a=1
#pragma unroll
    for (int j = 0; j < 8; ++j) {
        float b0 = bias[o * 32      + j + 8 * g];
        float b1 = bias[o * 32 + 16 + j + 8 * g];
        acc00[j] = b0; acc01[j] = b0; acc02[j] = b0; acc03[j] = b0;
        acc10[j] = b1; acc11[j] = b1; acc12[j] = b1; acc13[j] = b1;
    }

#pragma unroll
    for (int kh = 0; kh < 3; ++kh) {
#pragma unroll
        for (int kw = 0; kw < 3; ++kw) {
            const int t = kh * 3 + kw;
            // A lane base: W[oc = o*32 + l16][K = c0 + 2g (+1)]
            const float* wp = &wlds[(t * OUT_C + o * 32 + l16) * CSTR + 2 * g];
            // B lane base: X[row = 2*rp + kh][col = l16 + kw][K = c0 + 2g (+1)]
            const float* ip = &ilds[((2 * rp + kh) * ICOLS + l16 + kw) * CSTR + 2 * g];
#pragma unroll
            for (int c0 = 0; c0 < IN_C; c0 += 4) {
                v2f a0, a1, b00, b01, b10, b11;
                a0.x  = wp[c0];                         a0.y  = wp[c0 + 1];
                a1.x  = wp[16*CSTR + c0];               a1.y  = wp[16*CSTR + c0 + 1];
                b00.x = ip[c0];                         b00.y = ip[c0 + 1];
                b01.x = ip[16*CSTR + c0];               b01.y = ip[16*CSTR + c0 + 1];
                b10.x = ip[ICOLS*CSTR + c0];            b10.y = ip[ICOLS*CSTR + c0 + 1];
                b11.x = ip[(ICOLS+16)*CSTR + c0];       b11.y = ip[(ICOLS+16)*CSTR + c0 + 1];
                acc00 = WMMA_F32X4(a0, b00, acc00);
                acc01 = WMMA_F32X4(a0, b01, acc01);
                acc02 = WMMA_F32X4(a0, b10, acc02);
                acc03 = WMMA_F32X4(a0, b11, acc03);
                acc10 = WMMA_F32X4(a1, b00, acc10);
                acc11 = WMMA_F32X4(a1, b01, acc11);
                acc12 = WMMA_F32X4(a1, b10, acc12);
                acc13 = WMMA_F32X4(a1, b11, acc13);
            }
        }
    }

    // ---- store 32oc x (2 rows x 32w) tile (coalesced along w per half-wave)
    const int hA = h0 + 2 * rp;
    const int hB = hA + 1;
    const int wA = w0 + l16;
    const int wB = wA + 16;
#pragma unroll
    for (int j = 0; j < 8; ++j) {
        int oc0 = o * 32 + j + 8 * g;
        int oc1 = oc0 + 16;
        float* p0 = &out[((n * OUT_C + oc0) * HH) * WW];
        float* p1 = &out[((n * OUT_C + oc1) * HH) * WW];
        p0[hA * WW + wA] = acc00[j];
        p0[hA * WW + wB] = acc01[j];
        p0[hB * WW + wA] = acc02[j];
        p0[hB * WW + wB] = acc03[j];
        p1[hA * WW + wA] = acc10[j];
        p1[hA * WW + wB] = acc11[j];
        p1[hB * WW + wA] = acc12[j];
        p1[hB * WW + wB] = acc13[j];
    }
}

extern "C" void kernel_launch(void* const* d_in, const int* in_sizes, int n_in,
                              void* d_out, int out_size, void* d_ws, size_t ws_size,
                              hipStream_t stream) {
    const float* x    = (const float*)d_in[0];
    const float* wgt  = (const float*)d_in[1];
    const float* bias = (const float*)d_in[2];
    float* out        = (float*)d_out;

    const int    nblocks = NB * (HH / ROWS_PER_WG) * 7;               // 3136 workgroups
    const size_t shmem   = (size_t)(WLDS_F + ILDS_F) * sizeof(float); // ~121 KB (CDNA5: 320 KB/WGP)

    conv2d_wmma_f32_kernel<<<nblocks, 256, shmem, stream>>>(x, wgt, bias, out);
}